// QuadraticInhibitorAttention_55817394979295
// MI455X (gfx1250) — compile-verified
//
#include <hip/hip_runtime.h>

typedef __attribute__((ext_vector_type(2))) float v2f;
typedef __attribute__((ext_vector_type(8))) float v8f;
typedef __attribute__((ext_vector_type(4))) unsigned int v4u;
typedef __attribute__((ext_vector_type(8))) int v8i;
typedef __attribute__((ext_vector_type(4))) int v4i;

constexpr int kS  = 512;   // sequence length
constexpr int kH  = 512;   // hidden
constexpr int kNH = 8;     // heads
constexpr int kD  = 64;    // head dim
constexpr int kZP = kS + 4;  // padded z row stride (kill 512-stride bank conflicts)

// LDS layout (floats) for inhibitor kernel: v_s must be at offset 0 (TDM lds_addr).
constexpr int kVS_OFF  = 0;                      // [kS][kD] = 32768 floats
constexpr int kZS_OFF  = kS * kD;                // [16][kZP] = 8256 floats
constexpr int kACC_OFF = kZS_OFF + 16 * kZP;     // [16][kD] = 1024 floats
constexpr int kLDS_FLOATS = kACC_OFF + 16 * kD;  // total

// ---------------------------------------------------------------------------
// Y[m,n] = sum_c X[m,c] * W[n,c] + b[n]   (i.e. Y = X @ W^T + b)
// One wave32 per 16x16 output tile, fp32 WMMA, K stepped by 4.
// fp32 fragment layout (ISA 7.12.2):
//   A 16x4 : lanes 0-15 rows M=0..15, VGPR0=K(2*hi), VGPR1=K(2*hi+1)
//   B 4x16 : lanes 0-15 cols N=0..15, VGPR0=K(2*hi), VGPR1=K(2*hi+1)
//   C/D    : VGPR v holds (M = v + 8*hi, N = lane%16)
// ---------------------------------------------------------------------------
__global__ void __launch_bounds__(32)
gemm_xwt_kernel(const float* __restrict__ X, const float* __restrict__ W,
                const float* __restrict__ b, float* __restrict__ Y) {
  const int m0   = blockIdx.x * 16;
  const int n0   = blockIdx.y * 16;
  const int lane = threadIdx.x & 31;
  const int hi   = lane >> 4;
  const int lr   = lane & 15;

  const float bias = b[n0 + lr];
  v8f c;
#pragma unroll
  for (int i = 0; i < 8; ++i) c[i] = bias;

  const float* xrow = X + (size_t)(m0 + lr) * kH + 2 * hi;
  const float* wrow = W + (size_t)(n0 + lr) * kH + 2 * hi;

  for (int k0 = 0; k0 < kH; k0 += 4) {
    v2f a  = *(const v2f*)(xrow + k0);
    v2f bb = *(const v2f*)(wrow + k0);
    c = __builtin_amdgcn_wmma_f32_16x16x4_f32(false, a, false, bb,
                                              (short)0, c, false, false);
  }

  float* yout = Y + (size_t)(m0 + 8 * hi) * kH + n0 + lr;
#pragma unroll
  for (int i = 0; i < 8; ++i) yout[(size_t)i * kH] = c[i];
}

// ---------------------------------------------------------------------------
// q2[h*S+s] = sum_d q[s, h*64+d]^2 ; same for k2.
// ---------------------------------------------------------------------------
__global__ void __launch_bounds__(256)
sumsq_kernel(const float* __restrict__ q, const float* __restrict__ k,
             float* __restrict__ q2, float* __restrict__ k2) {
  const int idx = blockIdx.x * 256 + threadIdx.x;  // = h*kS + s
  const int h = idx >> 9;
  const int s = idx & (kS - 1);
  const float* qp = q + (size_t)s * kH + h * kD;
  const float* kp = k + (size_t)s * kH + h * kD;
  float sq = 0.f, sk = 0.f;
#pragma unroll 8
  for (int d = 0; d < kD; ++d) {
    sq += qp[d] * qp[d];
    sk += kp[d] * kp[d];
  }
  q2[idx] = sq;
  k2[idx] = sk;
}

// ---------------------------------------------------------------------------
// Per (head, 16-row i-tile), block = 1024 threads = 32 waves:
//  Phase 0: wave 0 issues a TDM tensor_load_to_lds DMA of the whole V head
//           panel (512 x 64 fp32, row stride 512 floats) into LDS (v_s),
//           tracked by TENSORcnt. (Fallback: cooperative VGPR copy.)
//  Phase 1: each wave computes one 16x16 QK^T tile via fp32 WMMA, converts to
//           z = coef*(|q|^2+|k|^2-2qk)+zc+mask and writes to padded z_s.
//  Phase 2: thread owns (i, 4 consecutive d) and 1/4 of the j range;
//           V read as ds_load_b128 (conflict-free), z as float4;
//           partials combined with ds_add_f32 atomics into acc_s.
// ---------------------------------------------------------------------------
__global__ void __launch_bounds__(1024)
inhibitor_kernel(const float* __restrict__ q, const float* __restrict__ k,
                 const float* __restrict__ v, const float* __restrict__ q2,
                 const float* __restrict__ k2, const float* __restrict__ mask,
                 const float* __restrict__ gamma, float* __restrict__ ctx) {
  extern __shared__ float smem[];
  float* v_s   = smem + kVS_OFF;   // [kS][kD], row-major == global tile layout
  float* z_s   = smem + kZS_OFF;   // [16][kZP]
  float* acc_s = smem + kACC_OFF;  // [16][kD]

  const int i0  = blockIdx.x * 16;
  const int h   = blockIdx.y;
  const int tid = threadIdx.x;

  const float g      = gamma[0];
  const float sqrt_d = 8.0f;  // sqrt(64)
  const float coef   = 15.0f / (16.0f * g * sqrt_d);
  const float zc     = (3.0f * (float)kD / 16.0f) / (g * sqrt_d);

  // Zero the accumulator tile (ordering guaranteed by the phase-1 barrier).
  for (int t = tid; t < 16 * kD; t += 1024) acc_s[t] = 0.0f;

  // ---- Phase 0: stage V head panel into LDS ----
#if __has_builtin(__builtin_amdgcn_tensor_load_to_lds) && \
    __has_builtin(__builtin_amdgcn_s_wait_tensorcnt)
  if (tid < 32) {
    const unsigned long long ga =
        (unsigned long long)(const void*)(v + (size_t)h * kD);
    v4u g0;
    g0[0] = 1u;                                   // count=1 (valid, user mode)
    g0[1] = 0u;                                   // lds_addr: v_s at LDS offset 0
    g0[2] = (unsigned)(ga & 0xFFFFFFFFu);         // global_addr[31:0]
    g0[3] = (unsigned)((ga >> 32) & 0x1FFFFFFu)   // global_addr[56:32]
            | (2u << 30);                         // type = 2 (image)
    v8i g1;
    g1[0] = (int)(2u << 16);    // wg_mask=0 | data_size=2 (4 bytes) | no flags
    g1[1] = (int)(64u << 16);   // atomic_barrier_addr=0 | tensor_dim0 = 64
    g1[2] = (int)(512u << 16);  // tensor_dim0 hi=0 | tensor_dim1 = 512
    g1[3] = (int)(64u << 16);   // tensor_dim1 hi=0 | tile_dim0 = 64
    g1[4] = (int)512;           // tile_dim1 = 512 | tile_dim2 = 0
    g1[5] = (int)512;           // tensor_dim0_stride = 512 (lo)
    g1[6] = 0;                  // stride hi | tensor_dim1_stride lo (unused, 2D)
    g1[7] = 0;
    v4i g2 = {0, 0, 0, 0};
    v4i g3 = {0, 0, 0, 0};
    v8i g4 = {0, 0, 0, 0, 0, 0, 0, 0};  // extra operand of 6-arg builtin
    __builtin_amdgcn_tensor_load_to_lds(g0, g1, g2, g3, g4, 0);
  }
#else
  for (int t = tid; t < kS * kD; t += 1024) {
    const int j = t >> 6;
    const int d = t & (kD - 1);
    v_s[t] = v[(size_t)j * kH + h * kD + d];
  }
#endif

  // ---- Phase 1: wave w computes the 16x16 qk tile for j0 = 16*w ----
  const int wave = tid >> 5;
  const int lane = tid & 31;
  const int hi   = lane >> 4;
  const int lr   = lane & 15;
  const int j0   = wave * 16;

  v8f c = {};
  const float* qrow = q + (size_t)(i0 + lr) * kH + h * kD + 2 * hi;
  const float* krow = k + (size_t)(j0 + lr) * kH + h * kD + 2 * hi;
#pragma unroll
  for (int d0 = 0; d0 < kD; d0 += 4) {
    v2f a  = *(const v2f*)(qrow + d0);
    v2f bb = *(const v2f*)(krow + d0);
    c = __builtin_amdgcn_wmma_f32_16x16x4_f32(false, a, false, bb,
                                              (short)0, c, false, false);
  }

  const int   j   = j0 + lr;
  const float k2j = k2[h * kS + j];
  const float pen = (1.0f - mask[j]) * 1e10f;
#pragma unroll
  for (int vv = 0; vv < 8; ++vv) {
    const int M   = vv + 8 * hi;
    const float z = coef * (q2[h * kS + i0 + M] + k2j - 2.0f * c[vv]) + zc + pen;
    z_s[M * kZP + j] = z;
  }

#if __has_builtin(__builtin_amdgcn_tensor_load_to_lds) && \
    __has_builtin(__builtin_amdgcn_s_wait_tensorcnt)
  if (tid < 32) __builtin_amdgcn_s_wait_tensorcnt(0);
#endif
  __syncthreads();

  // ---- Phase 2: relu-reduction over j, LDS-fed, b128 vector loads ----
  const int chunk = tid >> 8;            // 0..3 -> j range [128*chunk, +128)
  const int pair  = tid & 255;           // 0..255
  const int il    = pair >> 4;           // 0..15
  const int d4    = (pair & 15) << 2;    // 0,4,...,60
  const float* zp = z_s + il * kZP;
  const float* vp = v_s + d4;

  float a0 = 0.f, a1 = 0.f, a2 = 0.f, a3 = 0.f;
  const int jbeg = chunk * 128, jend = jbeg + 128;
#pragma unroll 4
  for (int jj = jbeg; jj < jend; jj += 4) {
    const float4 z4 = *(const float4*)(zp + jj);
    const float4 w0 = *(const float4*)(vp + (size_t)(jj + 0) * kD);
    const float4 w1 = *(const float4*)(vp + (size_t)(jj + 1) * kD);
    const float4 w2 = *(const float4*)(vp + (size_t)(jj + 2) * kD);
    const float4 w3 = *(const float4*)(vp + (size_t)(jj + 3) * kD);
    a0 += fmaxf(w0.x - z4.x, 0.f) + fmaxf(w1.x - z4.y, 0.f) +
          fmaxf(w2.x - z4.z, 0.f) + fmaxf(w3.x - z4.w, 0.f);
    a1 += fmaxf(w0.y - z4.x, 0.f) + fmaxf(w1.y - z4.y, 0.f) +
          fmaxf(w2.y - z4.z, 0.f) + fmaxf(w3.y - z4.w, 0.f);
    a2 += fmaxf(w0.z - z4.x, 0.f) + fmaxf(w1.z - z4.y, 0.f) +
          fmaxf(w2.z - z4.z, 0.f) + fmaxf(w3.z - z4.w, 0.f);
    a3 += fmaxf(w0.w - z4.x, 0.f) + fmaxf(w1.w - z4.y, 0.f) +
          fmaxf(w2.w - z4.z, 0.f) + fmaxf(w3.w - z4.w, 0.f);
  }
  atomicAdd(&acc_s[il * kD + d4 + 0], a0);
  atomicAdd(&acc_s[il * kD + d4 + 1], a1);
  atomicAdd(&acc_s[il * kD + d4 + 2], a2);
  atomicAdd(&acc_s[il * kD + d4 + 3], a3);
  __syncthreads();

  // ---- Writeout: one element per thread ----
  const int wi = tid >> 6;        // 0..15
  const int wd = tid & (kD - 1);  // 0..63
  ctx[(size_t)(i0 + wi) * kH + h * kD + wd] = acc_s[wi * kD + wd];
}

// ---------------------------------------------------------------------------
extern "C" void kernel_launch(void* const* d_in, const int* in_sizes, int n_in,
                              void* d_out, int out_size, void* d_ws, size_t ws_size,
                              hipStream_t stream) {
  const float* hs    = (const float*)d_in[0];
  const float* mask  = (const float*)d_in[1];
  const float* Wq    = (const float*)d_in[2];
  const float* bq    = (const float*)d_in[3];
  const float* Wk    = (const float*)d_in[4];
  const float* bk    = (const float*)d_in[5];
  const float* Wv    = (const float*)d_in[6];
  const float* bv    = (const float*)d_in[7];
  const float* Wo    = (const float*)d_in[8];
  const float* bo    = (const float*)d_in[9];
  const float* gamma = (const float*)d_in[10];
  float* out = (float*)d_out;

  float* ws  = (float*)d_ws;
  float* q   = ws;
  float* k   = ws + (size_t)kS * kH;
  float* v   = ws + 2 * (size_t)kS * kH;
  float* ctx = ws + 3 * (size_t)kS * kH;
  float* q2  = ws + 4 * (size_t)kS * kH;
  float* k2  = q2 + kNH * kS;

  const dim3 gt(kS / 16, kH / 16);
  gemm_xwt_kernel<<<gt, 32, 0, stream>>>(hs, Wq, bq, q);
  gemm_xwt_kernel<<<gt, 32, 0, stream>>>(hs, Wk, bk, k);
  gemm_xwt_kernel<<<gt, 32, 0, stream>>>(hs, Wv, bv, v);

  sumsq_kernel<<<(kNH * kS) / 256, 256, 0, stream>>>(q, k, q2, k2);

  const size_t lds_bytes = (size_t)kLDS_FLOATS * sizeof(float);
  (void)hipFuncSetAttribute((const void*)inhibitor_kernel,
                            hipFuncAttributeMaxDynamicSharedMemorySize,
                            (int)lds_bytes);
  inhibitor_kernel<<<dim3(kS / 16, kNH), 1024, lds_bytes, stream>>>(
      q, k, v, q2, k2, mask, gamma, ctx);

  gemm_xwt_kernel<<<gt, 32, 0, stream>>>(ctx, Wo, bo, out);
}